// AUAvULoss_23184233464523
// MI455X (gfx1250) — compile-verified
//
#include <hip/hip_runtime.h>
#include <hip/hip_bf16.h>
#include <stdint.h>

#define WAVES   8
#define CP      1024          // padded LDS row stride (floats), 16B-aligned rows
#define EPSF    1e-12f

typedef int v4i __attribute__((ext_vector_type(4)));
typedef __attribute__((address_space(1))) v4i* gv4p;   // global v4i*
typedef __attribute__((address_space(3))) v4i* lv4p;   // LDS v4i*

// ---------------------------------------------------------------------------
// CDNA5 async global->LDS staging (16-byte chunk per lane)
// ---------------------------------------------------------------------------
__device__ __forceinline__ void async_copy16(const float* g, float* l) {
#if defined(__gfx1250__) && __has_builtin(__builtin_amdgcn_global_load_async_to_lds_b128)
    __builtin_amdgcn_global_load_async_to_lds_b128(
        (gv4p)(uintptr_t)g, (lv4p)(uintptr_t)l, 0, 0);
#elif defined(__gfx1250__)
    unsigned loff = (unsigned)(uintptr_t)l;        // low 32 bits == LDS byte offset
    unsigned long long ga = (unsigned long long)(uintptr_t)g;
    asm volatile("global_load_async_to_lds_b128 %0, %1, off"
                 :: "v"(loff), "v"(ga) : "memory");
#else
    ((float4*)l)[0] = ((const float4*)g)[0];       // host pass / other arch
#endif
}

__device__ __forceinline__ void wait_async0() {
#if defined(__gfx1250__) && __has_builtin(__builtin_amdgcn_s_wait_asynccnt)
    __builtin_amdgcn_s_wait_asynccnt(0);
    asm volatile("" ::: "memory");
#elif defined(__gfx1250__)
    asm volatile("s_wait_asynccnt 0x0" ::: "memory");
#endif
}

// Stage C floats from global row into LDS row via async b128 loads.
__device__ __forceinline__ void stage_row(const float* grow, float* lrow, int lane, int C) {
    const int nfull = C / 128;                 // full 32-lane x 16B rounds
    const int tailv = (C - nfull * 128) / 4;   // remaining 16B chunks (0..31)
    for (int r = 0; r < nfull; ++r) {
        const int e = (r * 32 + lane) * 4;
        async_copy16(grow + e, lrow + e);
    }
    if (lane < tailv) {
        const int e = nfull * 128 + lane * 4;
        async_copy16(grow + e, lrow + e);
    }
    const int rem = C & 3;
    if (lane < rem) {                          // scalar leftovers (none for C=1000)
        const int e = C - rem + lane;
        lrow[e] = grow[e];
    }
    asm volatile("" ::: "memory");
}

// ---------------------------------------------------------------------------
// Kernel 1: per-row softmax stats. One wave (32 lanes) per row.
//   max/argmax pass + fused sumexp/entropy pass:
//   H = log S - (1/S) * sum e_j*(x_j - m)   (avoids per-element log)
// ---------------------------------------------------------------------------
__global__ __launch_bounds__(256) void k_rowstats(
    const float* __restrict__ logits, const int* __restrict__ labels,
    int N, int C, int rowsPerBlock,
    float* __restrict__ unc, float* __restrict__ conf, float* __restrict__ accA,
    float* __restrict__ cePart, float* __restrict__ minPart, float* __restrict__ maxPart)
{
    __shared__ float lbuf[WAVES * CP];
    __shared__ float red[3][WAVES];

    const int tid  = threadIdx.x;
    const int wave = tid >> 5;
    const int lane = tid & 31;
    const int base0 = blockIdx.x * rowsPerBlock;

    float wce = 0.f, wmin = 3.0e38f, wmax = -3.0e38f;

    for (int it = 0; it < rowsPerBlock; it += WAVES) {
        const int row = base0 + it + wave;
        const bool valid = (it + wave) < rowsPerBlock && row < N;   // wave-uniform
        if (valid) {
            const float* grow = logits + (size_t)row * (size_t)C;
            float* lrow = &lbuf[wave * CP];
            const float4* lrow4 = (const float4*)lrow;
            const int C4 = C >> 2;

            stage_row(grow, lrow, lane, C);
            wait_async0();

            // ---- pass 1: max + argmax (first-max tie rule), float4 chunks
            float m = -3.0e38f; int am = 0x7fffffff;
            for (int c = lane; c < C4; c += 32) {
                const float4 x = lrow4[c];
                const int b = c << 2;
                if (x.x > m) { m = x.x; am = b;     }
                if (x.y > m) { m = x.y; am = b + 1; }
                if (x.z > m) { m = x.z; am = b + 2; }
                if (x.w > m) { m = x.w; am = b + 3; }
            }
            for (int j = (C4 << 2) + lane; j < C; j += 32) {  // scalar remainder
                const float x = lrow[j];
                if (x > m) { m = x; am = j; }
            }
            for (int off = 16; off; off >>= 1) {
                const float m2 = __shfl_xor(m, off, 32);
                const int   a2 = __shfl_xor(am, off, 32);
                if (m2 > m || (m2 == m && a2 < am)) { m = m2; am = a2; }
            }

            // ---- pass 2: fused  s = sum e,  t = sum e*(x-m)
            float s = 0.f, t = 0.f;
            for (int c = lane; c < C4; c += 32) {
                const float4 x = lrow4[c];
                const float d0 = x.x - m, d1 = x.y - m, d2 = x.z - m, d3 = x.w - m;
                const float e0 = __expf(d0), e1 = __expf(d1);
                const float e2 = __expf(d2), e3 = __expf(d3);
                s += (e0 + e1) + (e2 + e3);
                t += e0 * d0 + e1 * d1 + e2 * d2 + e3 * d3;
            }
            for (int j = (C4 << 2) + lane; j < C; j += 32) {  // scalar remainder
                const float d = lrow[j] - m;
                const float e = __expf(d);
                s += e; t += e * d;
            }
            for (int off = 16; off; off >>= 1) {
                s += __shfl_xor(s, off, 32);
                t += __shfl_xor(t, off, 32);
            }

            if (lane == 0) {
                const float invs = 1.f / s;
                const float logS = __logf(s);
                const float u = logS - t * invs;        // entropy
                const int lab = labels[row];
                const float xl = lrow[lab];             // logit at label
                unc[row]  = u;
                conf[row] = invs;                       // max prob = exp(0)/S
                accA[row] = (am == lab) ? 1.f : 0.f;
                wce += xl - m - logS;                   // log-softmax at label
                wmin = fminf(wmin, u);
                wmax = fmaxf(wmax, u);
            }
        }
    }

    if (lane == 0) { red[0][wave] = wce; red[1][wave] = wmin; red[2][wave] = wmax; }
    __syncthreads();
    if (tid == 0) {
        float c = 0.f, mn = 3.0e38f, mx = -3.0e38f;
        for (int w = 0; w < WAVES; ++w) {
            c += red[0][w]; mn = fminf(mn, red[1][w]); mx = fmaxf(mx, red[2][w]);
        }
        cePart[blockIdx.x] = c; minPart[blockIdx.x] = mn; maxPart[blockIdx.x] = mx;
    }
}

// ---------------------------------------------------------------------------
// Kernel 2: fold per-block partials -> scalars; zero the diff-histogram.
// scal: [0]=umin [1]=umax [2]=ce_sum
// ---------------------------------------------------------------------------
__global__ __launch_bounds__(256) void k_reduce(
    const float* __restrict__ cePart, const float* __restrict__ minPart,
    const float* __restrict__ maxPart, int G,
    float* __restrict__ scal, float* __restrict__ hist)
{
    __shared__ float rc[256], rn[256], rx[256];
    const int t = threadIdx.x;
    float c = 0.f, mn = 3.0e38f, mx = -3.0e38f;
    for (int i = t; i < G; i += 256) {
        c += cePart[i]; mn = fminf(mn, minPart[i]); mx = fmaxf(mx, maxPart[i]);
    }
    rc[t] = c; rn[t] = mn; rx[t] = mx;
    __syncthreads();
    for (int s = 128; s; s >>= 1) {
        if (t < s) {
            rc[t] += rc[t + s];
            rn[t] = fminf(rn[t], rn[t + s]);
            rx[t] = fmaxf(rx[t], rx[t + s]);
        }
        __syncthreads();
    }
    if (t == 0) { scal[0] = rn[0]; scal[1] = rx[0]; scal[2] = rc[0]; }
    if (t < 88) hist[t] = 0.f;
}

// ---------------------------------------------------------------------------
// Kernel 3: per-row threshold pass using difference arrays.
// hist layout: [0..21]=dAC  [22..43]=dAU  [44..65]=dIC  [66..87]=dIU
// certain set per row is a suffix [k0,20] -> only 2 LDS atomics per row.
// ---------------------------------------------------------------------------
__global__ __launch_bounds__(256) void k_thresh(
    const float* __restrict__ unc, const float* __restrict__ conf,
    const float* __restrict__ accA, int N,
    const float* __restrict__ scal, float* __restrict__ hist)
{
    __shared__ float lh[88];
    const int t = threadIdx.x;
    if (t < 88) lh[t] = 0.f;
    __syncthreads();

    const int i = blockIdx.x * blockDim.x + t;
    const float umin = scal[0], umax = scal[1];
    const float range = umax - umin;

    if (i < N) {
        const float u  = unc[i];
        const float cf = conf[i];
        const bool acc = accA[i] != 0.f;
        const float tu = tanhf(u);
        const float base = acc ? cf : (1.f - cf);
        const float vc = base * (1.f - tu);   // value on the "certain" side
        const float vu = base * tu;           // value on the "uncertain" side
        int k0 = 0;                            // # thresholds with u > th_k
        for (int k = 0; k < 21; ++k) {
            const float th = umin + ((float)k / 20.f) * range;
            k0 += (u > th) ? 1 : 0;
        }
        const int off = acc ? 0 : 44;
        atomicAdd(&lh[off + k0], vc);
        atomicAdd(&lh[off + 22 + k0], vu);
    }
    __syncthreads();
    if (t < 88) atomicAdd(&hist[t], lh[t]);
}

// ---------------------------------------------------------------------------
// Kernel 4: prefix/suffix sums -> AvU curve -> trapezoid AUC -> loss.
// ---------------------------------------------------------------------------
__global__ void k_final(const float* __restrict__ hist,
                        const float* __restrict__ scal,
                        int N, float* __restrict__ out)
{
    if (threadIdx.x != 0 || blockIdx.x != 0) return;

    float nac[21], nau[21], nic[21], niu[21];
    float pAC = 0.f, pIC = 0.f;
    for (int k = 0; k < 21; ++k) {
        pAC += hist[k];      nac[k] = pAC;
        pIC += hist[44 + k]; nic[k] = pIC;
    }
    float totAU = 0.f, totIU = 0.f;
    for (int j = 0; j < 22; ++j) { totAU += hist[22 + j]; totIU += hist[66 + j]; }
    float cAU = 0.f, cIU = 0.f;
    for (int k = 0; k < 21; ++k) {
        cAU += hist[22 + k]; nau[k] = totAU - cAU;
        cIU += hist[66 + k]; niu[k] = totIU - cIU;
    }

    float auc = 0.f, prev = 0.f;
    for (int k = 0; k < 21; ++k) {
        const float num = nac[k] + niu[k];
        const float den = nac[k] + nau[k] + nic[k] + niu[k] + EPSF;
        const float avu = num / den;
        if (k > 0) {
            const float dth = ((float)k / 20.f) - ((float)(k - 1) / 20.f);
            auc += 0.5f * (avu + prev) * dth;
        }
        prev = avu;
    }
    const float ce = -scal[2] / (float)N;
    out[0] = -3.0f * __logf(auc + EPSF) + ce;
}

// ---------------------------------------------------------------------------
// Launcher
// ---------------------------------------------------------------------------
extern "C" void kernel_launch(void* const* d_in, const int* in_sizes, int n_in,
                              void* d_out, int out_size, void* d_ws, size_t ws_size,
                              hipStream_t stream)
{
    const float* logits = (const float*)d_in[0];
    const int*   labels = (const int*)d_in[1];
    const int N = in_sizes[1];
    const int C = in_sizes[0] / N;

    // workspace layout (floats): unc[N] conf[N] acc[N] cePart[G1] minPart[G1]
    //                            maxPart[G1] scal[4] hist[88]   (~800 KB)
    const int G1 = 1024;
    float* ws      = (float*)d_ws;
    float* unc     = ws;
    float* conf    = unc + N;
    float* accA    = conf + N;
    float* cePart  = accA + N;
    float* minPart = cePart + G1;
    float* maxPart = minPart + G1;
    float* scal    = maxPart + G1;
    float* hist    = scal + 4;

    const int rowsPerBlock = (N + G1 - 1) / G1;

    k_rowstats<<<G1, 256, 0, stream>>>(logits, labels, N, C, rowsPerBlock,
                                       unc, conf, accA, cePart, minPart, maxPart);
    k_reduce<<<1, 256, 0, stream>>>(cePart, minPart, maxPart, G1, scal, hist);

    const int G3 = (N + 255) / 256;
    k_thresh<<<G3, 256, 0, stream>>>(unc, conf, accA, N, scal, hist);
    k_final<<<1, 32, 0, stream>>>(hist, scal, N, (float*)d_out);
}